// KalmanFilter_8564164788768
// MI455X (gfx1250) — compile-verified
//
#include <hip/hip_runtime.h>
#include <hip/hip_bf16.h>

// Kalman filter + RTS smoother, MI455X (gfx1250, wave32).
// One wave per batch (512 waves). All per-step 32x32 linear algebra is kept in
// LDS; every O(N^3) product runs on V_WMMA_F32_16X16X4_F32 (f32 precision is
// required for covariance recursions). The backward pass recomputes Sig_p /
// mu_p from the stored filtered results, so no global workspace is needed:
// forward writes mu_f/Sig_f into the output buffers, backward overwrites them
// in place with the smoothed values.

#define BBq 512
#define Tq  128
#define Nq  32
#define Mq  16
#define Pq  16

typedef __attribute__((ext_vector_type(2))) float v2f;
typedef __attribute__((ext_vector_type(8))) float v8f;

#define WSYNC() __builtin_amdgcn_wave_barrier()

__device__ __forceinline__ v8f wmma4(v2f a, v2f b, v8f c) {
  // D = A(16x4) * B(4x16) + C(16x16), all f32.
  return __builtin_amdgcn_wmma_f32_16x16x4_f32(
      /*neg_a=*/false, a, /*neg_b=*/false, b,
      /*c_mod=*/(short)0, c, /*reuse_a=*/false, /*reuse_b=*/false);
}

__device__ __forceinline__ float mat_el(const float* b, int ld, bool tr, int r, int c) {
  return tr ? b[c * ld + r] : b[r * ld + c];
}

// A-operand fragment (16x4 f32): VGPR0 = K {0 | 2}, VGPR1 = K {1 | 3} per lane half.
__device__ __forceinline__ v2f lda_frag(const float* X, int ld, bool tr,
                                        int m0, int k0, int lane) {
  int h = lane >> 4, m = m0 + (lane & 15);
  v2f a;
  a.x = mat_el(X, ld, tr, m, k0 + 2 * h);
  a.y = mat_el(X, ld, tr, m, k0 + 2 * h + 1);
  return a;
}

// B-operand fragment (4x16 f32): VGPR v holds rows {v, v+2} split across lane halves.
__device__ __forceinline__ v2f ldb_frag(const float* Y, int ld, bool tr,
                                        int k0, int n0, int lane) {
  int h = lane >> 4, n = n0 + (lane & 15);
  v2f b;
  b.x = mat_el(Y, ld, tr, k0 + 2 * h, n);
  b.y = mat_el(Y, ld, tr, k0 + 2 * h + 1, n);
  return b;
}

// C/D tile (16x16 f32): VGPR v holds rows {v, v+8} per lane half, col = lane&15.
__device__ __forceinline__ v8f ld_ctile(const float* D, int ld, int m0, int n0, int lane) {
  int h = lane >> 4, n = n0 + (lane & 15);
  v8f c;
#pragma unroll
  for (int v = 0; v < 8; ++v) c[v] = D[(m0 + v + 8 * h) * ld + n];
  return c;
}

__device__ __forceinline__ void st_ctile(float* D, int ld, int m0, int n0, int lane, v8f acc) {
  int h = lane >> 4, n = n0 + (lane & 15);
#pragma unroll
  for (int v = 0; v < 8; ++v) D[(m0 + v + 8 * h) * ld + n] = acc[v];
}

// Wave-level tiled GEMM on LDS:  D[Mt*16 x Nt*16] (+)= X * Y, K = Kt*16.
// trx/tryb read the operand transposed (stride-swapped) for free A^T/B^T.
__device__ __forceinline__ void wave_mm(float* D, int ldd,
                                        const float* X, int ldx, bool trx,
                                        const float* Y, int ldy, bool tryb,
                                        int Mt, int Nt, int Kt, int lane, bool acc) {
  for (int mt = 0; mt < Mt; ++mt) {
    for (int nt = 0; nt < Nt; ++nt) {
      v8f c;
      if (acc) {
        c = ld_ctile(D, ldd, mt * 16, nt * 16, lane);
      } else {
#pragma unroll
        for (int v = 0; v < 8; ++v) c[v] = 0.0f;
      }
      for (int kt = 0; kt < Kt; ++kt) {
#pragma unroll
        for (int kk = 0; kk < 4; ++kk) {
          v2f a = lda_frag(X, ldx, trx, mt * 16, kt * 16 + kk * 4, lane);
          v2f b = ldb_frag(Y, ldy, tryb, kt * 16 + kk * 4, nt * 16, lane);
          c = wmma4(a, b, c);
        }
      }
      st_ctile(D, ldd, mt * 16, nt * 16, lane, c);
    }
  }
  WSYNC();
}

// Wave-cooperative Gauss-Jordan inverse of SPD n x n (n = 16 or 32).
// Augmented [S | I] lives in `aug` (row stride w); lane owns columns
// {lane, lane+32}. Inverse ends up at aug[:, n:2n]. No pivoting (SPD).
template <int n>
__device__ __forceinline__ void gj_invert(float* aug, int w, const float* S,
                                          int lds, int lane) {
  constexpr int w2 = 2 * n;
  for (int c = lane; c < w2; c += 32)
    for (int i = 0; i < n; ++i)
      aug[i * w + c] = (c < n) ? S[i * lds + c] : (((c - n) == i) ? 1.0f : 0.0f);
  WSYNC();
  for (int p = 0; p < n; ++p) {
    float ip = 1.0f / aug[p * w + p];
    WSYNC();
    for (int c = lane; c < w2; c += 32) aug[p * w + c] *= ip;
    WSYNC();
    float pr0 = aug[p * w + lane];
    float pr1 = (w2 > 32) ? aug[p * w + lane + 32] : 0.0f;
    for (int r = 0; r < n; ++r) {
      if (r == p) continue;
      float f = aug[r * w + p];           // lockstep: all lanes load before stores
      aug[r * w + lane] -= f * pr0;
      if (w2 > 32) aug[r * w + lane + 32] -= f * pr1;
    }
    WSYNC();
  }
}

__global__ __launch_bounds__(32) void kalman_smooth_kernel(
    const float* __restrict__ Yg, const float* __restrict__ Ug,
    const float* __restrict__ Ag, const float* __restrict__ Bg,
    const float* __restrict__ Cg, const float* __restrict__ mu0g,
    const float* __restrict__ Sig0g, const float* __restrict__ Qg,
    const float* __restrict__ Rg,
    float* __restrict__ outMu, float* __restrict__ outSig) {
  // Padded strides (33/17/65) vs the 64 LDS banks. ~49 KB total per wave.
  __shared__ float sA[32 * 33], sQ[32 * 33], sBm[32 * 17], sCm[16 * 33], sR[16 * 17];
  __shared__ float Sig[32 * 33], SP[32 * 33], T1[32 * 33], T2[32 * 33], SF[32 * 33];
  __shared__ float AUG[32 * 65], KM[32 * 17], PCT[32 * 17], Ssm[16 * 17];
  __shared__ float vmu[32], vmup[32], vms[32], vmuf[32], uv[16], yv[16], rv[16];

  const int lane = threadIdx.x;  // single wave per workgroup
  const int b = blockIdx.x;

  // Stage shared constants into LDS (once per 2*128 recursion steps).
  for (int i = lane; i < 32 * 32; i += 32) sA[(i >> 5) * 33 + (i & 31)] = Ag[i];
  for (int i = lane; i < 32 * 32; i += 32) sQ[(i >> 5) * 33 + (i & 31)] = Qg[i];
  for (int i = lane; i < 32 * 16; i += 32) sBm[(i >> 4) * 17 + (i & 15)] = Bg[i];
  for (int i = lane; i < 16 * 32; i += 32) sCm[(i >> 5) * 33 + (i & 31)] = Cg[i];
  for (int i = lane; i < 16 * 16; i += 32) sR[(i >> 4) * 17 + (i & 15)] = Rg[i];
  vmu[lane] = mu0g[lane];
  for (int i = 0; i < 32; ++i) Sig[i * 33 + lane] = Sig0g[i * 32 + lane];
  WSYNC();

  const float* Yb = Yg + (size_t)b * Tq * Pq;
  const float* Ub = Ug + (size_t)b * Tq * Mq;
  float* muB = outMu + (size_t)b * Tq * Nq;
  float* sgB = outSig + (size_t)b * Tq * Nq * Nq;

  // ---------------- forward Kalman filter ----------------
  for (int t = 0; t < Tq; ++t) {
    if (lane < 16) { uv[lane] = Ub[t * 16 + lane]; yv[lane] = Yb[t * 16 + lane]; }
    WSYNC();
    // mu_p = A mu + B u
    {
      float s = 0.f;
      for (int j = 0; j < 32; ++j) s += sA[lane * 33 + j] * vmu[j];
      for (int j = 0; j < 16; ++j) s += sBm[lane * 17 + j] * uv[j];
      vmup[lane] = s;
    }
    WSYNC();
    // Sig_p = A Sig A^T + Q
    wave_mm(T1, 33, sA, 33, false, Sig, 33, false, 2, 2, 2, lane, false);
    wave_mm(SP, 33, T1, 33, false, sA, 33, true, 2, 2, 2, lane, false);
    for (int i = 0; i < 32; ++i) SP[i * 33 + lane] += sQ[i * 33 + lane];
    WSYNC();
    // PCT = Sig_p C^T ; S = sym(C PCT + R)
    wave_mm(PCT, 17, SP, 33, false, sCm, 33, true, 2, 1, 2, lane, false);
    wave_mm(Ssm, 17, sCm, 33, false, PCT, 17, false, 1, 1, 2, lane, false);
    if (lane < 16) {
      float col[16];
      for (int i = 0; i < 16; ++i)
        col[i] = 0.5f * (Ssm[i * 17 + lane] + Ssm[lane * 17 + i]) + sR[i * 17 + lane];
      for (int i = 0; i < 16; ++i) Ssm[i * 17 + lane] = col[i];
    }
    WSYNC();
    gj_invert<16>(AUG, 33, Ssm, 17, lane);
    // K = PCT S^-1
    wave_mm(KM, 17, PCT, 17, false, AUG + 16, 33, false, 2, 1, 1, lane, false);
    // r = y - C mu_p
    if (lane < 16) {
      float s = yv[lane];
      for (int j = 0; j < 32; ++j) s -= sCm[lane * 33 + j] * vmup[j];
      rv[lane] = s;
    }
    WSYNC();
    // mu_f = mu_p + K r  (store to output; backward reads it back)
    {
      float s = vmup[lane];
      for (int j = 0; j < 16; ++j) s += KM[lane * 17 + j] * rv[j];
      vmu[lane] = s;
      muB[t * 32 + lane] = s;
    }
    WSYNC();
    // Joseph form: Sig_f = (I-KC) Sig_p (I-KC)^T + K R K^T
    wave_mm(T1, 33, KM, 17, false, sCm, 33, false, 2, 2, 1, lane, false);
    for (int i = 0; i < 32; ++i)
      T2[i * 33 + lane] = ((i == lane) ? 1.0f : 0.0f) - T1[i * 33 + lane];
    WSYNC();
    wave_mm(T1, 33, T2, 33, false, SP, 33, false, 2, 2, 2, lane, false);
    wave_mm(Sig, 33, T1, 33, false, T2, 33, true, 2, 2, 2, lane, false);
    wave_mm(PCT, 17, KM, 17, false, sR, 17, false, 2, 1, 1, lane, false);
    wave_mm(Sig, 33, PCT, 17, false, KM, 17, true, 2, 2, 1, lane, true);
    // symmetrize, keep as carry, spill Sig_f to output slot
    for (int i = 0; i < 32; ++i)
      T1[i * 33 + lane] = 0.5f * (Sig[i * 33 + lane] + Sig[lane * 33 + i]);
    WSYNC();
    for (int i = 0; i < 32; ++i) {
      float v = T1[i * 33 + lane];
      Sig[i * 33 + lane] = v;
      sgB[t * 1024 + i * 32 + lane] = v;
    }
    WSYNC();
  }

  // ---------------- backward RTS smoother ----------------
  // Carry = (mu_f[T-1], Sig_f[T-1]) already in LDS and in the output buffers.
  vms[lane] = vmu[lane];
  WSYNC();
  for (int t = Tq - 2; t >= 0; --t) {
    for (int i = 0; i < 32; ++i) SF[i * 33 + lane] = sgB[t * 1024 + i * 32 + lane];
    vmuf[lane] = muB[t * 32 + lane];
    if (lane < 16) uv[lane] = Ub[(t + 1) * 16 + lane];
    WSYNC();
    // Recompute Sig_p[t+1] = A Sig_f[t] A^T + Q ; mu_p[t+1] = A mu_f + B u[t+1]
    wave_mm(T1, 33, sA, 33, false, SF, 33, false, 2, 2, 2, lane, false);
    wave_mm(SP, 33, T1, 33, false, sA, 33, true, 2, 2, 2, lane, false);
    for (int i = 0; i < 32; ++i) SP[i * 33 + lane] += sQ[i * 33 + lane];
    {
      float s = 0.f;
      for (int j = 0; j < 32; ++j) s += sA[lane * 33 + j] * vmuf[j];
      for (int j = 0; j < 16; ++j) s += sBm[lane * 17 + j] * uv[j];
      vmup[lane] = s;
    }
    WSYNC();
    gj_invert<32>(AUG, 65, SP, 33, lane);
    // J = Sig_f A^T Sig_p^-1
    wave_mm(T1, 33, SF, 33, false, sA, 33, true, 2, 2, 2, lane, false);
    wave_mm(T2, 33, T1, 33, false, AUG + 32, 65, false, 2, 2, 2, lane, false);
    // Dm = Sig_s[t+1] - Sig_p (reuse SP)
    for (int i = 0; i < 32; ++i) SP[i * 33 + lane] = Sig[i * 33 + lane] - SP[i * 33 + lane];
    WSYNC();
    wave_mm(T1, 33, T2, 33, false, SP, 33, false, 2, 2, 2, lane, false);
    wave_mm(SF, 33, T1, 33, false, T2, 33, true, 2, 2, 2, lane, true);  // SF += J Dm J^T
    // mu_s = mu_f + J (mu_s[t+1] - mu_p)
    {
      float s = vmuf[lane];
      for (int j = 0; j < 32; ++j) s += T2[lane * 33 + j] * (vms[j] - vmup[j]);
      WSYNC();
      vms[lane] = s;
      muB[t * 32 + lane] = s;
    }
    WSYNC();
    // symmetrize -> new carry + output
    for (int i = 0; i < 32; ++i)
      T1[i * 33 + lane] = 0.5f * (SF[i * 33 + lane] + SF[lane * 33 + i]);
    WSYNC();
    for (int i = 0; i < 32; ++i) {
      float v = T1[i * 33 + lane];
      Sig[i * 33 + lane] = v;
      sgB[t * 1024 + i * 32 + lane] = v;
    }
    WSYNC();
  }
}

extern "C" void kernel_launch(void* const* d_in, const int* in_sizes, int n_in,
                              void* d_out, int out_size, void* d_ws, size_t ws_size,
                              hipStream_t stream) {
  (void)in_sizes; (void)n_in; (void)out_size; (void)d_ws; (void)ws_size;
  const float* Y   = (const float*)d_in[0];
  const float* U   = (const float*)d_in[1];
  const float* A   = (const float*)d_in[2];
  const float* Bm  = (const float*)d_in[3];
  const float* C   = (const float*)d_in[4];
  const float* mu0 = (const float*)d_in[5];
  const float* S0  = (const float*)d_in[6];
  const float* Q   = (const float*)d_in[7];
  const float* R   = (const float*)d_in[8];
  float* out = (float*)d_out;
  float* outMu  = out;                                  // (BB, T, N)
  float* outSig = out + (size_t)BBq * Tq * Nq;          // (BB, T, N, N)
  kalman_smooth_kernel<<<BBq, 32, 0, stream>>>(Y, U, A, Bm, C, mu0, S0, Q, R,
                                               outMu, outSig);
}